// All2AllDenseEmbedding_28080496181534
// MI455X (gfx1250) — compile-verified
//
#include <hip/hip_runtime.h>
#include <hip/hip_bf16.h>

// All2AllDenseEmbedding: out[row, :] = table[idx[row], :]
//   idx:   [16384*26*10] int64   (values in [0, 1e6) -> low 32 bits suffice)
//   table: [1'000'000, 32] float32  (128 MB -> resident in MI455X 192 MB L2)
//   out:   [16384*26*10, 32] float32 (545 MB, streamed with NT stores)
//
// Pure gather, zero FLOPs -> bandwidth-bound (~30 us roofline at 23.3 TB/s).
// Strategy:
//   * 128-bit loads/stores (global_load_b128 / global_store_b128)
//   * NT stores: the 545 MB output stream must not evict the table from L2
//   * RT (default) table loads so the table stays L2-resident
//   * load only the low 4 bytes of each int64 index (b32)
//   * unconditional speculative global_prefetch_b8 on the index stream
//   * uniform, check-free main loop (exact grid coverage) unrolled x2 for
//     two independent gathers in flight per wave; 8 waves/block x 16,640
//     blocks for massive latency hiding.

typedef float f4 __attribute__((ext_vector_type(4)));

#define EMB_VEC 8            // 32 floats = 8 x float4 per embedding row
#define THREADS 256
#define ITERS_PER_THREAD 8

__global__ __launch_bounds__(THREADS)
void All2AllDenseEmbedding_gather_kernel(const unsigned int* __restrict__ idx_lo32,
                                         const f4* __restrict__ table,
                                         f4* __restrict__ out,
                                         unsigned int per_thread,
                                         unsigned int n_chunks) {
    const unsigned int stride = gridDim.x * blockDim.x;
    unsigned int i = blockIdx.x * blockDim.x + threadIdx.x;

    // Main loop: every thread runs exactly per_thread iterations, no bounds
    // check -> trip count is wave-uniform and the body is branch-free, so the
    // unroller can interleave independent gathers.
#pragma unroll 2
    for (unsigned int k = 0; k < per_thread; ++k, i += stride) {
        // Speculative prefetch of the next iteration's index cacheline.
        // OOB speculative prefetches are silently dropped by hardware.
        __builtin_prefetch(&idx_lo32[(size_t)((i + stride) >> 3) * 2], 0, 3);

        const unsigned int row = i >> 3;   // which lookup
        const unsigned int sub = i & 7;    // which float4 within the 128B row

        // int64 index, little-endian: low 32 bits at element 2*row.
        const unsigned int e = idx_lo32[(size_t)row * 2];

        // 128-bit gather load from the (L2-resident) table, RT hint.
        f4 v = table[(size_t)e * EMB_VEC + sub];

        // 128-bit streaming store, NT hint.
        __builtin_nontemporal_store(v, &out[(size_t)i]);
    }

    // Tail (never taken for the fixed problem size, kept for generality).
    if (i < n_chunks) {
        const unsigned int e = idx_lo32[(size_t)(i >> 3) * 2];
        f4 v = table[(size_t)e * EMB_VEC + (i & 7)];
        __builtin_nontemporal_store(v, &out[(size_t)i]);
    }
}

extern "C" void kernel_launch(void* const* d_in, const int* in_sizes, int n_in,
                              void* d_out, int out_size, void* d_ws, size_t ws_size,
                              hipStream_t stream) {
    // setup_inputs() order: inputs (int64 [16384,26,10]), table (f32 [1e6,32]).
    const unsigned int* idx_lo32 = (const unsigned int*)d_in[0]; // int64 viewed as 2x u32
    const f4*           table    = (const f4*)d_in[1];
    f4*                 out      = (f4*)d_out;

    const unsigned int n_rows   = (unsigned int)in_sizes[0];   // 4,259,840
    const unsigned int n_chunks = n_rows * EMB_VEC;            // 34,078,720

    // Choose grid so stride divides n_chunks exactly for the fixed problem
    // size: 34,078,720 = 16,640 blocks * 256 threads * 8 iters.
    const unsigned int total_threads =
        (n_chunks + ITERS_PER_THREAD - 1) / ITERS_PER_THREAD;
    const unsigned int blocks = (total_threads + THREADS - 1) / THREADS; // 16,640
    const unsigned int stride = blocks * THREADS;
    const unsigned int per_thread = n_chunks / stride;         // 8 (uniform part)

    All2AllDenseEmbedding_gather_kernel<<<blocks, THREADS, 0, stream>>>(
        idx_lo32, table, out, per_thread, n_chunks);
}